// GreedyDecoderUpdate_14525579395594
// MI455X (gfx1250) — compile-verified
//
#include <hip/hip_runtime.h>
#include <hip/hip_bf16.h>
#include <stdint.h>

// ---------------- problem constants ----------------
#define B_SZ 1024
#define T_SZ 128
#define H_ENC 1024
#define L_SZ 2
#define H_PRED 320
#define MAXLEN 256
#define BLANK 28
#define MAX_SYMBOLS 30

// ---------------- output layout (float elements, concatenated in return order)
#define O_SYM   0
#define O_RES   (O_SYM  + B_SZ)               // 1024
#define O_RIDX  (O_RES  + B_SZ*MAXLEN)        // 263168
#define O_TIDX  (O_RIDX + B_SZ)               // 264192
#define O_FI    (O_TIDX + B_SZ)               // 265216
#define O_PREG  (O_FI   + B_SZ*H_ENC)         // 1313792
#define O_PHG   (O_PREG + B_SZ)               // 1314816
#define O_PCG   (O_PHG  + L_SZ*B_SZ*H_PRED)   // 1970176
#define O_DONE  (O_PCG  + L_SZ*B_SZ*H_PRED)   // 2625536

typedef float f4 __attribute__((ext_vector_type(4)));
typedef int   i4 __attribute__((ext_vector_type(4)));

#define AS1 __attribute__((address_space(1)))
#define AS3 __attribute__((address_space(3)))

#if __has_builtin(__builtin_amdgcn_global_load_async_to_lds_b128) && \
    __has_builtin(__builtin_amdgcn_global_store_async_from_lds_b128)
#define HAVE_ASYNC_LDS 1
#else
#define HAVE_ASYNC_LDS 0
#endif

__device__ __forceinline__ f4 ldnt_f4(const float* p) {
  return __builtin_nontemporal_load((const f4*)p);
}
__device__ __forceinline__ void stnt_f4(float* p, f4 v) {
  __builtin_nontemporal_store(v, (f4*)p);
}
__device__ __forceinline__ bool calc_nb(int s, int a, int t, int fl) {
  return (s != BLANK) & (a < MAX_SYMBOLS) & (t < fl);
}

// ---------------- kernel 0: init done flag ----------------
__global__ void k_init(float* out) { out[O_DONE] = 1.0f; }

// ---------------- kernel 1: per-batch scalar state ----------------
__global__ void k_scalar(const int* __restrict__ sym, const int* __restrict__ sa,
                         const int* __restrict__ ridx, const int* __restrict__ ti,
                         const int* __restrict__ fl, const int* __restrict__ preg,
                         float* __restrict__ out) {
  int b = blockIdx.x * blockDim.x + threadIdx.x;
  if (b >= B_SZ) return;
  int s = sym[b], a = sa[b], t = ti[b], fle = fl[b];
  bool nb = calc_nb(s, a, t, fle);
  int t_new = nb ? t : (t + 1);
  out[O_SYM + b]  = (float)(nb ? (a + 1) : 0);
  out[O_RIDX + b] = (float)(ridx[b] + (nb ? 1 : 0));
  out[O_TIDX + b] = (float)t_new;
  out[O_PREG + b] = (float)(nb ? s : preg[b]);
  if (t_new < fle) out[O_DONE] = 0.0f;   // racing same-value stores: deterministic
}

// ---------------- kernel 2: res row update + int->float ----------------
// one block per batch row, 64 threads x int4 (256 ints per row)
__global__ void k_res(const int* __restrict__ sym, const int* __restrict__ sa,
                      const int* __restrict__ ti, const int* __restrict__ fl,
                      const int* __restrict__ ridx, const int* __restrict__ res,
                      float* __restrict__ out) {
  int b = blockIdx.x;
  int s = sym[b];
  bool nb = calc_nb(s, sa[b], ti[b], fl[b]);
  int r = ridx[b];
  int j = threadIdx.x * 4;
  i4 v = __builtin_nontemporal_load((const i4*)(res + b * MAXLEN + j));
  f4 o;
#pragma unroll
  for (int k = 0; k < 4; ++k) {
    int val = v[k];
    if (nb && (j + k) == r) val = s;
    o[k] = (float)val;
  }
  stnt_f4(out + O_RES + b * MAXLEN + j, o);
}

// ---------------- kernel 3: fi_new select-gather copy (4 KB/row) ----------
// one block per batch, 256 threads x 16B. Uses CDNA5 async global<->LDS DMA
// (ASYNCcnt path): global_load_async_to_lds_b128 / global_store_async_from_lds_b128.
__global__ void k_fi(const int* __restrict__ sym, const int* __restrict__ sa,
                     const int* __restrict__ ti, const int* __restrict__ fl,
                     const float* __restrict__ fi, const float* __restrict__ f,
                     float* __restrict__ out) {
  int b = blockIdx.x;
  int s = sym[b];
  int t = ti[b];
  bool nb = calc_nb(s, sa[b], t, fl[b]);
  int t_new = nb ? t : (t + 1);
  int t_g = t_new < 0 ? 0 : (t_new > (T_SZ - 1) ? (T_SZ - 1) : t_new);
  const float* src = nb ? (fi + (size_t)b * H_ENC)
                        : (f + ((size_t)b * T_SZ + (size_t)t_g) * H_ENC);
  float* dst = out + O_FI + (size_t)b * H_ENC;
  int tid = threadIdx.x;

#if HAVE_ASYNC_LDS
  __shared__ char lbuf[256 * 16];
  AS3 char* lp = (AS3 char*)lbuf;
  AS1 char* gs = (AS1 char*)src;
  AS1 char* gd = (AS1 char*)dst;
  // DMA 16B/lane: global -> LDS (ASYNCcnt++)
  __builtin_amdgcn_global_load_async_to_lds_b128((AS1 i4*)(gs + tid * 16),
                                                 (AS3 i4*)(lp + tid * 16), 0, 0);
#if __has_builtin(__builtin_amdgcn_s_wait_asynccnt)
  __builtin_amdgcn_s_wait_asynccnt(0);
#else
  asm volatile("s_wait_asynccnt 0" ::: "memory");
#endif
  // DMA 16B/lane: LDS -> global (wave-private region, no barrier needed)
  __builtin_amdgcn_global_store_async_from_lds_b128((AS1 i4*)(gd + tid * 16),
                                                    (AS3 i4*)(lp + tid * 16), 0, 0);
  // S_ENDPGM performs implicit wait-idle, so outstanding ASYNCcnt drains.
#else
  f4 v = ldnt_f4(src + tid * 4);
  stnt_f4(dst + tid * 4, v);
#endif
}

// ---------------- kernel 4: pre_hg / pre_cg masked copy -------------------
// one block per batch; 320 threads = 10 waves, branchless pointer selects.
// tid < 160 -> hg tensor, tid >= 160 -> cg tensor (wave-aligned split).
__global__ void k_pred(const int* __restrict__ sym, const int* __restrict__ sa,
                       const int* __restrict__ ti, const int* __restrict__ fl,
                       const float* __restrict__ pre_hg, const float* __restrict__ pre_cg,
                       const float* __restrict__ hg, const float* __restrict__ cg,
                       float* __restrict__ out) {
  int b = blockIdx.x;
  bool nb = calc_nb(sym[b], sa[b], ti[b], fl[b]);
  int tid = threadIdx.x;           // 0..319
  int tensor = tid >= 160;         // 0: hg, 1: cg
  int r = tid - tensor * 160;      // 0..159
  int l = r >= 80;                 // layer
  int j = r - l * 80;              // float4 index within row (80 per row)
  size_t off = ((size_t)l * B_SZ + (size_t)b) * H_PRED + (size_t)j * 4;
  const float* srcA = tensor ? cg : hg;
  const float* srcB = tensor ? pre_cg : pre_hg;
  const float* src = nb ? srcA : srcB;
  float* dst = out + (tensor ? O_PCG : O_PHG) + off;
  f4 v = ldnt_f4(src + off);
  stnt_f4(dst, v);
}

extern "C" void kernel_launch(void* const* d_in, const int* in_sizes, int n_in,
                              void* d_out, int out_size, void* d_ws, size_t ws_size,
                              hipStream_t stream) {
  const int*   symbols       = (const int*)d_in[0];
  const int*   symbols_added = (const int*)d_in[1];
  const int*   res           = (const int*)d_in[2];
  const int*   res_idx       = (const int*)d_in[3];
  const float* f             = (const float*)d_in[4];
  const int*   f_lens        = (const int*)d_in[5];
  const int*   time_idx      = (const int*)d_in[6];
  const float* fi            = (const float*)d_in[7];
  const int*   pre_g         = (const int*)d_in[8];
  const float* pre_hg        = (const float*)d_in[9];
  const float* pre_cg        = (const float*)d_in[10];
  const float* hg            = (const float*)d_in[11];
  const float* cg            = (const float*)d_in[12];
  float* out = (float*)d_out;

  k_init<<<1, 1, 0, stream>>>(out);
  k_scalar<<<B_SZ / 256, 256, 0, stream>>>(symbols, symbols_added, res_idx,
                                           time_idx, f_lens, pre_g, out);
  k_res<<<B_SZ, 64, 0, stream>>>(symbols, symbols_added, time_idx, f_lens,
                                 res_idx, res, out);
  k_fi<<<B_SZ, 256, 0, stream>>>(symbols, symbols_added, time_idx, f_lens,
                                 fi, f, out);
  k_pred<<<B_SZ, 320, 0, stream>>>(symbols, symbols_added, time_idx, f_lens,
                                   pre_hg, pre_cg, hg, cg, out);
}